// Forward_Warp_65730179498100
// MI455X (gfx1250) — compile-verified
//
#include <hip/hip_runtime.h>
#include <stdint.h>

#define BB 8
#define CC 16
#define HH 512
#define WW 512
#define HW (HH * WW)
#define TILE_W 256

typedef __attribute__((ext_vector_type(4))) unsigned u32x4;
typedef __attribute__((ext_vector_type(8))) unsigned u32x8;

// Zero-initialize output with wide 128-bit stores (harness poisons d_out).
__global__ __launch_bounds__(256) void fw_zero_kernel(float4* __restrict__ out, int n4) {
    int i = blockIdx.x * blockDim.x + threadIdx.x;
    if (i < n4) out[i] = make_float4(0.f, 0.f, 0.f, 0.f);
}

__global__ __launch_bounds__(256) void Forward_Warp_65730179498100_kernel(
    const float* __restrict__ im0,
    const float2* __restrict__ flow,
    float* __restrict__ out) {
    // 16 channels x 256 pixels of im0 staged by the Tensor Data Mover.
    __shared__ float tile[CC * TILE_W];

    const int w = blockIdx.x * TILE_W + threadIdx.x;
    const int h = blockIdx.y;
    const int b = blockIdx.z;

    // ---- TDM: DMA the [16 x 256] f32 tile (row stride = H*W) into LDS ----
    // Issue from wave 0 only (condition is wave-uniform -> execz skip on waves 1..7).
    // Do NOT wait here: the DMA overlaps with the flow load + weight math below.
    if ((threadIdx.x >> 5) == 0) {
        uint64_t gaddr = (uint64_t)(uintptr_t)(im0 + (size_t)(b * CC) * HW
                                               + (size_t)h * WW
                                               + (size_t)blockIdx.x * TILE_W);
        uint32_t lds_off = (uint32_t)(uintptr_t)&tile[0];

        u32x4 g0;
        g0[0] = 1u;                                   // count=1 (valid), user mode
        g0[1] = lds_off;                              // lds_addr [63:32]
        g0[2] = (uint32_t)gaddr;                      // global_addr lo
        g0[3] = (uint32_t)(gaddr >> 32) | (2u << 30); // global_addr hi | type=2 (image)

        u32x8 g1;
        g1[0] = (2u << 16);                // wg_mask=0 | data_size=2 (4 bytes)
        g1[1] = ((uint32_t)TILE_W) << 16;  // tensor_dim0 low16 @ bits 48..63
        g1[2] = ((uint32_t)CC) << 16;      // tensor_dim0 hi=0 | tensor_dim1 low16
        g1[3] = ((uint32_t)TILE_W) << 16;  // tensor_dim1 hi=0 | tile_dim0=256
        g1[4] = (uint32_t)CC;              // tile_dim1=16 | tile_dim2=0
        g1[5] = (uint32_t)HW;              // tensor_dim0_stride low32 (262144 elems)
        g1[6] = 0u;                        // dim0_stride hi | dim1_stride lo (unused, 2D)
        g1[7] = 0u;

        asm volatile("tensor_load_to_lds %0, %1" :: "s"(g0), "s"(g1) : "memory");
    }

    // ---- Bilinear splat weights (matches reference math exactly) ----
    // Runs on all waves while the TDM streams the im0 tile into LDS.
    float2 f = flow[((size_t)b * HH + h) * WW + w];
    float x = (float)w + f.x;
    float y = (float)h + f.y;
    float x0f = floorf(x);
    float y0f = floorf(y);
    int xi0 = (int)x0f, yi0 = (int)y0f;
    int xi1 = xi0 + 1,  yi1 = yi0 + 1;
    float ax = x - x0f;        // (x - x0)
    float ay = y - y0f;        // (y - y0)
    float bx = 1.0f - ax;      // (x1 - x)
    float by = 1.0f - ay;      // (y1 - y)

    float wt[4]  = { bx * by, ax * by, bx * ay, ax * ay }; // nw, ne, sw, se
    int   xs[4]  = { xi0, xi1, xi0, xi1 };
    int   ys[4]  = { yi0, yi0, yi1, yi1 };
    int   flat[4];
#pragma unroll
    for (int k = 0; k < 4; ++k) {
        bool valid = (xs[k] >= 0) && (xs[k] < WW) && (ys[k] >= 0) && (ys[k] < HH);
        int xc = min(max(xs[k], 0), WW - 1);
        int yc = min(max(ys[k], 0), HH - 1);
        flat[k] = yc * WW + xc;
        wt[k]   = valid ? wt[k] : 0.0f;   // reference adds 0 at clamped index when OOB
    }

    // Now require the tile: wave 0 drains its TENSORcnt (no-op for waves 1..7),
    // then the workgroup barrier makes the LDS tile visible to everyone.
    __builtin_amdgcn_s_wait_tensorcnt(0);
    __syncthreads();

    // ---- Scatter: 16 channels x 4 corners of global f32 atomics (L2-resident) ----
#pragma unroll
    for (int c = 0; c < CC; ++c) {
        float v = tile[c * TILE_W + threadIdx.x];
        float* outc = out + (size_t)(b * CC + c) * HW;
#pragma unroll
        for (int k = 0; k < 4; ++k) {
            atomicAdd(outc + flat[k], v * wt[k]);
        }
    }
}

extern "C" void kernel_launch(void* const* d_in, const int* in_sizes, int n_in,
                              void* d_out, int out_size, void* d_ws, size_t ws_size,
                              hipStream_t stream) {
    (void)in_sizes; (void)n_in; (void)d_ws; (void)ws_size;
    const float*  im0  = (const float*)d_in[0];
    const float2* flow = (const float2*)d_in[1];
    float*        out  = (float*)d_out;

    // Zero-init output (33,554,432 floats -> 8,388,608 float4).
    int n4 = out_size / 4;
    fw_zero_kernel<<<(n4 + 255) / 256, 256, 0, stream>>>((float4*)out, n4);

    // One block per (256-pixel row segment, row, batch).
    dim3 grid(WW / TILE_W, HH, BB);
    Forward_Warp_65730179498100_kernel<<<grid, 256, 0, stream>>>(im0, flow, out);
}